// Att_Stack_21973052686617
// MI455X (gfx1250) — compile-verified
//
#include <hip/hip_runtime.h>

typedef __attribute__((ext_vector_type(2))) float v2f;
typedef __attribute__((ext_vector_type(8))) float v8f;

#define BT    4              // batch elements per wave/workgroup
#define NSEQ  30
#define VROWS (BT * NSEQ)    // 120 valid rows
#define PROWS 128            // padded to 8 WMMA m-tiles
#define MTILES (PROWS / 16)

struct AttP { const float *wq,*bq,*wk,*bk,*wv,*bv,*wm1,*bm1,*wm2,*bm2; };
struct ParamsAll {
  AttP a1, a2, a3;
  const float *n1a,*n1b,*n2a,*n2b,*n3a,*n3b;
  const float *fwm1,*fbm1,*fwm2,*fbm2;
};

// leaky_relu(x, 0.01) == max(x, 0.01x): one mul + one max, no cmp/cndmask.
__device__ inline float leaky(float x) { return fmaxf(x, 0.01f * x); }

// Stage a (DOUT x DIN) row-major weight matrix into LDS at rows [rowoff, rowoff+DOUT),
// row stride DP = DIN padded to multiple of 4. Pad regions may hold stale data:
// pad-K cols multiply zero A values, pad-N rows feed unstored C columns.
template<int DIN, int DOUT>
__device__ void stage_w(const float* W, float* WB, int rowoff, int lane)
{
  constexpr int DP = (DIN + 3) & ~3;
  for (int i = lane; i < DOUT * DIN; i += 32)
    WB[(rowoff + i / DIN) * DP + (i % DIN)] = W[i];
}

// OUT[rows][DOUT] = IN[rows][DIN] @ W^T + bias via v_wmma_f32_16x16x4_f32.
// A frag: lane M = lane&15, VGPR j holds K = 4*ks + j + 2*(lane>>4)   (ISA 7.12.2)
// B frag: same K striping, N = lane&15.  C frag: VGPR v -> row v + 8*(lane>>4), col = lane&15.
// All fragment loads are unconditional ds_load_b64 (activations/weights zero-padded).
template<int DIN, int DOUT, int INS, int OUTS, bool LK>
__device__ void linear_wmma(const float* in, float* out,
                            const float* W, const float* bias,
                            float* WB, float* WBias, int lane)
{
  constexpr int DP = (DIN + 3) & ~3;
  constexpr int KS = DP / 4;
  constexpr int NT = (DOUT + 15) / 16;

  stage_w<DIN, DOUT>(W, WB, 0, lane);
  if (lane < DOUT) WBias[lane] = bias[lane];
  __syncthreads();

  const int half = lane >> 4;
  const int lc   = lane & 15;

  v2f  Breg[NT][KS];
  float bb[NT];
#pragma unroll
  for (int nt = 0; nt < NT; ++nt) {
    const int col = nt * 16 + lc;
    bb[nt] = WBias[col];                       // stale beyond DOUT: never stored
#pragma unroll
    for (int ks = 0; ks < KS; ++ks)
      Breg[nt][ks] = *(const v2f*)(WB + col * DP + ks * 4 + 2 * half);
  }

  for (int mt = 0; mt < MTILES; ++mt) {
    v8f zero = {};
    v8f acc[NT];
#pragma unroll
    for (int nt = 0; nt < NT; ++nt) acc[nt] = zero;

    const float* arow = in + (mt * 16 + lc) * INS;
#pragma unroll
    for (int ks = 0; ks < KS; ++ks) {
      v2f a = *(const v2f*)(arow + ks * 4 + 2 * half);
#pragma unroll
      for (int nt = 0; nt < NT; ++nt)
        acc[nt] = __builtin_amdgcn_wmma_f32_16x16x4_f32(
            false, a, false, Breg[nt][ks], (short)0, acc[nt], false, false);
    }

#pragma unroll
    for (int nt = 0; nt < NT; ++nt) {
      const int col = nt * 16 + lc;
      if (col < DOUT) {
#pragma unroll
        for (int v = 0; v < 8; ++v) {
          const int row = mt * 16 + v + 8 * half;
          float r = acc[nt][v] + bb[nt];
          if (LK) r = leaky(r);
          out[row * OUTS + col] = r;
        }
      }
    }
  }
  __syncthreads();   // before WB is re-staged by the next layer
}

// Fused Q/K/V projection: one A-fragment load feeds three WMMAs.
template<int DIN>
__device__ void qkv_wmma(const AttP& p, const float* in,
                         float* Qb, float* Kb, float* Vb,
                         float* WB, float* WBias, int lane)
{
  constexpr int DP = (DIN + 3) & ~3;
  constexpr int KS = DP / 4;

  stage_w<DIN, 10>(p.wq, WB,  0, lane);
  stage_w<DIN, 10>(p.wk, WB, 16, lane);
  stage_w<DIN, 10>(p.wv, WB, 32, lane);
  if (lane < 10) {
    WBias[lane]      = p.bq[lane];
    WBias[16 + lane] = p.bk[lane];
    WBias[32 + lane] = p.bv[lane];
  }
  __syncthreads();

  const int half = lane >> 4;
  const int lc   = lane & 15;

  v2f  Breg[3][KS];
  float bb[3];
#pragma unroll
  for (int t = 0; t < 3; ++t) {
    bb[t] = WBias[t * 16 + lc];
#pragma unroll
    for (int ks = 0; ks < KS; ++ks)
      Breg[t][ks] = *(const v2f*)(WB + (t * 16 + lc) * DP + ks * 4 + 2 * half);
  }

  float* const outs[3] = {Qb, Kb, Vb};
  for (int mt = 0; mt < MTILES; ++mt) {
    v8f zero = {};
    v8f acc[3] = {zero, zero, zero};

    const float* arow = in + (mt * 16 + lc) * 20;
#pragma unroll
    for (int ks = 0; ks < KS; ++ks) {
      v2f a = *(const v2f*)(arow + ks * 4 + 2 * half);
#pragma unroll
      for (int t = 0; t < 3; ++t)
        acc[t] = __builtin_amdgcn_wmma_f32_16x16x4_f32(
            false, a, false, Breg[t][ks], (short)0, acc[t], false, false);
    }

    if (lc < 10) {
#pragma unroll
      for (int t = 0; t < 3; ++t) {
#pragma unroll
        for (int v = 0; v < 8; ++v) {
          const int row = mt * 16 + v + 8 * half;
          outs[t][row * 12 + lc] = acc[t][v] + bb[t];
        }
      }
    }
  }
  __syncthreads();
}

// Causal sin-scored attention, online softmax, rows -> lanes.
// Both heads (H=2, DH=5) fused into one key loop: each K/V row loaded once as b64 pairs.
__device__ void attention(const float* Qb, const float* Kb, const float* Vb,
                          float* Cb, int lane)
{
  const float scale = 0.44721359549995793f;  // 1/sqrt(5)
  for (int pass = 0; pass < (VROWS + 31) / 32; ++pass) {
    const int r = pass * 32 + lane;
    if (r < VROWS) {
      const int b = r / NSEQ;
      const int n = r - b * NSEQ;
      const int kbase = b * NSEQ;

      float q[10];
#pragma unroll
      for (int d = 0; d < 10; d += 2) {
        v2f t = *(const v2f*)(Qb + r * 12 + d);
        q[d] = t[0]; q[d + 1] = t[1];
      }
      float m0 = -1e30f, m1 = -1e30f, l0 = 0.f, l1 = 0.f;
      float acc[10] = {};
      for (int j = 0; j <= n; ++j) {
        const int kr = (kbase + j) * 12;
        float kv[10], vv[10];
#pragma unroll
        for (int d = 0; d < 10; d += 2) {
          v2f t = *(const v2f*)(Kb + kr + d); kv[d] = t[0]; kv[d + 1] = t[1];
          v2f u = *(const v2f*)(Vb + kr + d); vv[d] = u[0]; vv[d + 1] = u[1];
        }
        float s0 = 0.f, s1 = 0.f;
#pragma unroll
        for (int d = 0; d < 5; ++d) {
          s0 += q[d] * kv[d];
          s1 += q[5 + d] * kv[5 + d];
        }
        s0 = __sinf(s0 * scale);
        s1 = __sinf(s1 * scale);
        const float mn0 = fmaxf(m0, s0), mn1 = fmaxf(m1, s1);
        const float c0 = __expf(m0 - mn0), c1 = __expf(m1 - mn1);
        const float p0 = __expf(s0 - mn0), p1 = __expf(s1 - mn1);
        l0 = l0 * c0 + p0;
        l1 = l1 * c1 + p1;
#pragma unroll
        for (int d = 0; d < 5; ++d) {
          acc[d]     = acc[d]     * c0 + p0 * vv[d];
          acc[5 + d] = acc[5 + d] * c1 + p1 * vv[5 + d];
        }
        m0 = mn0; m1 = mn1;
      }
      const float i0 = __builtin_amdgcn_rcpf(l0);
      const float i1 = __builtin_amdgcn_rcpf(l1);
#pragma unroll
      for (int d = 0; d < 5; ++d) {
        Cb[r * 12 + d]     = acc[d] * i0;
        Cb[r * 12 + 5 + d] = acc[5 + d] * i1;
      }
    }
  }
}

// y = alpha*(x - mu)/(std_ddof1 + 1e-6) + beta over 20 features, rows -> lanes.
__device__ void layernorm(const float* Xb, float* Tb,
                          const float* al, const float* be, int lane)
{
  for (int pass = 0; pass < (VROWS + 31) / 32; ++pass) {
    const int r = pass * 32 + lane;
    if (r < VROWS) {
      float v[20]; float s = 0.f;
#pragma unroll
      for (int c = 0; c < 20; ++c) { v[c] = Xb[r * 20 + c]; s += v[c]; }
      const float mu = s * (1.f / 20.f);
      float var = 0.f;
#pragma unroll
      for (int c = 0; c < 20; ++c) { const float d = v[c] - mu; var += d * d; }
      const float sd = sqrtf(var * (1.f / 19.f));
      const float inv = __builtin_amdgcn_rcpf(sd + 1e-6f);
#pragma unroll
      for (int c = 0; c < 20; ++c) Tb[r * 20 + c] = al[c] * (v[c] - mu) * inv + be[c];
    }
  }
}

template<int DIN>
__device__ void att_block(const AttP p, const float* IN, float* OUT,
                          float* Qb, float* Kb, float* Vb, float* Cb, float* Hb,
                          float* WB, float* WBias, int lane)
{
  qkv_wmma<DIN>(p, IN, Qb, Kb, Vb, WB, WBias, lane);
  attention(Qb, Kb, Vb, Cb, lane);
  __syncthreads();
  linear_wmma<10, 20, 12, 20, true >(Cb, Hb, p.wm1, p.bm1, WB, WBias, lane);  // leaky MLP1
  linear_wmma<20, 20, 20, 20, false>(Hb, OUT, p.wm2, p.bm2, WB, WBias, lane); // MLP2
}

__global__ __launch_bounds__(32)
void att_stack_kernel(const float* __restrict__ x, ParamsAll P,
                      float* __restrict__ out)
{
  __shared__ float lds[14832];          // ~59 KB, one wave32 per workgroup
  float* X   = lds;                     // 128*20 running activation
  float* T   = lds + 2560;              // 128*20 normed input / MLP hidden
  float* A2  = lds + 5120;              // 128*20 attention-block output
  float* Qb  = lds + 7680;              // 128*12
  float* Kb  = lds + 9216;              // 128*12
  float* Vb  = lds + 10752;             // 128*12
  float* Cb  = lds + 12288;             // 128*12 concat / final output
  float* WB  = lds + 13824;             // 48*20 staged padded weights
  float* WBs = lds + 14784;             // 48 staged bias
  const int  lane = threadIdx.x;
  const long base = (long)blockIdx.x * VROWS;   // flat (batch*seq) row base

  for (int i = lane; i < 14832; i += 32) lds[i] = 0.f;
  __syncthreads();

  // load x rows (2 floats each, 8B aligned) into T cols 0..1 (cols 2..3 stay zero)
  for (int i = lane; i < VROWS; i += 32) {
    v2f xx = *(const v2f*)(x + (base + i) * 2);
    *(v2f*)(T + i * 20) = xx;
  }
  __syncthreads();

  att_block<2>(P.a1, T, X, Qb, Kb, Vb, Cb, /*Hb=*/T, WB, WBs, lane);   // x1

  layernorm(X, T, P.n1a, P.n1b, lane); __syncthreads();
  att_block<20>(P.a2, T, A2, Qb, Kb, Vb, Cb, T, WB, WBs, lane);
  for (int i = lane; i < PROWS * 20; i += 32) X[i] += A2[i];           // x2
  __syncthreads();

  layernorm(X, T, P.n2a, P.n2b, lane); __syncthreads();
  att_block<20>(P.a3, T, A2, Qb, Kb, Vb, Cb, T, WB, WBs, lane);
  for (int i = lane; i < PROWS * 20; i += 32) X[i] += A2[i];           // x3
  __syncthreads();

  layernorm(X, T, P.n3a, P.n3b, lane); __syncthreads();
  linear_wmma<20, 20, 20, 20, true >(T, A2, P.fwm1, P.fbm1, WB, WBs, lane);
  linear_wmma<20,  2, 20, 12, false>(A2, Cb, P.fwm2, P.fbm2, WB, WBs, lane);

  for (int i = lane; i < VROWS; i += 32) {
    v2f o;
    o[0] = Cb[i * 12 + 0];                                  // mean
    o[1] = fminf(fmaxf(Cb[i * 12 + 1], -10.f), 10.f);       // clipped logvar
    *(v2f*)(out + (base + i) * 2) = o;                      // global_store_b64
  }
}

extern "C" void kernel_launch(void* const* d_in, const int* in_sizes, int n_in,
                              void* d_out, int out_size, void* d_ws, size_t ws_size,
                              hipStream_t stream)
{
  // setup_inputs() insertion order: x, then params pytree in insertion order:
  // att1{wq,bq,wk,bk,wv,bv,wm1,bm1,wm2,bm2}, att2{...}, att3{...},
  // norm1_a, norm1_b, norm2_a, norm2_b, norm3_a, norm3_b, wm1, bm1, wm2, bm2
  int i = 0;
  auto F = [&]() { return (const float*)d_in[i++]; };
  const float* x = F();
  ParamsAll P;
  auto att = [&](AttP& a) {
    a.wq = F(); a.bq = F(); a.wk = F(); a.bk = F(); a.wv = F(); a.bv = F();
    a.wm1 = F(); a.bm1 = F(); a.wm2 = F(); a.bm2 = F();
  };
  att(P.a1); att(P.a2); att(P.a3);
  P.n1a = F(); P.n1b = F(); P.n2a = F(); P.n2b = F(); P.n3a = F(); P.n3b = F();
  P.fwm1 = F(); P.fbm1 = F(); P.fwm2 = F(); P.fbm2 = F();

  const int nblocks = 16384 / BT;   // 4096 wave32 workgroups
  att_stack_kernel<<<nblocks, 32, 0, stream>>>(x, P, (float*)d_out);
}